// Normalizer_65498251264084
// MI455X (gfx1250) — compile-verified
//
#include <hip/hip_runtime.h>
#include <math.h>

typedef float v4f __attribute__((ext_vector_type(4)));
typedef int   v4i __attribute__((vector_size(16)));   // matches async-builtin's param type

#define NCOLS 1024   // row length (N)
#define BLK   256    // 8 wave32s; each thread owns one float4 of the row

// ds_swizzle group-of-32 immediates: and=0x1F, or=0, xor=mask<<10 (full-wave butterfly)
#if __has_builtin(__builtin_amdgcn_ds_swizzle)
#define SWZ_ADD(p, imm)                                                        \
    (p) += __int_as_float(__builtin_amdgcn_ds_swizzle(__float_as_int(p), (imm)))
#else
#define SWZ_ADD(p, imm) (p) += __shfl_xor((p), ((imm) >> 10) & 0x1F)
#endif

// One block normalizes one row: out[row, :] = row / sum(row)  (0 if 1/sum non-finite)
__global__ __launch_bounds__(BLK)
void rownorm_kernel(const float* __restrict__ src, float* __restrict__ dst) {
    __shared__ float row[NCOLS];     // 4 KB staging of the row
    __shared__ float wsum[BLK / 32]; // per-wave partials
    __shared__ float s_inv;

    const int    tid  = threadIdx.x;
    const size_t base = (size_t)blockIdx.x * NCOLS;
    const float* gsrc = src + base;

#if __has_builtin(__builtin_amdgcn_global_load_async_to_lds_b128)
    // CDNA5 async DMA: global -> LDS, 16B per lane, tracked by ASYNCcnt.
    {
        typedef __attribute__((address_space(1))) v4i* gvptr_t;
        typedef __attribute__((address_space(3))) v4i* lvptr_t;
        gvptr_t g = (gvptr_t)(gsrc + tid * 4);
        lvptr_t l = (lvptr_t)(row + tid * 4);
        __builtin_amdgcn_global_load_async_to_lds_b128(g, l, 0, 0);
    }
  #if __has_builtin(__builtin_amdgcn_s_wait_asynccnt)
    __builtin_amdgcn_s_wait_asynccnt(0);
  #else
    asm volatile("s_wait_asynccnt 0" ::: "memory");
  #endif
#else
    // Fallback: nontemporal 128-bit load, park in LDS ourselves.
    {
        v4f v = __builtin_nontemporal_load((const v4f*)gsrc + tid);
        *((v4f*)row + tid) = v;
    }
#endif
    __syncthreads();

    // Each thread's float4 of the row (from LDS).
    v4f v = *((v4f*)row + tid);
    float p = v.x + v.y + v.z + v.w;

    // wave32 butterfly reduction via ds_swizzle (1 DS op + 1 add per step)
    SWZ_ADD(p, 0x041F); // xor 1
    SWZ_ADD(p, 0x081F); // xor 2
    SWZ_ADD(p, 0x101F); // xor 4
    SWZ_ADD(p, 0x201F); // xor 8
    SWZ_ADD(p, 0x401F); // xor 16
    if ((tid & 31) == 0) wsum[tid >> 5] = p;
    __syncthreads();

    if (tid == 0) {
        float d = 0.0f;
#pragma unroll
        for (int i = 0; i < BLK / 32; ++i) d += wsum[i];
        float inv = 1.0f / d;
        s_inv = __builtin_isfinite(inv) ? inv : 0.0f;
    }
    __syncthreads();

    const float inv = s_inv;
    v4f o;
    o.x = v.x * inv;
    o.y = v.y * inv;
    o.z = v.z * inv;
    o.w = v.w * inv;
    // Write-once 256 MiB output: nontemporal store, don't thrash L2.
    __builtin_nontemporal_store(o, (v4f*)(dst + base) + tid);
}

extern "C" void kernel_launch(void* const* d_in, const int* in_sizes, int n_in,
                              void* d_out, int out_size, void* d_ws, size_t ws_size,
                              hipStream_t stream) {
    const float* a0 = (const float*)d_in[0];
    const float* a1 = (const float*)d_in[1];
    float* out = (float*)d_out;

    // in_sizes[0] == B*N*N ; rows per matrix = B*N
    const int    rows = in_sizes[0] / NCOLS;
    const size_t per  = (size_t)rows * NCOLS;

    rownorm_kernel<<<dim3(rows), dim3(BLK), 0, stream>>>(a0, out);
    rownorm_kernel<<<dim3(rows), dim3(BLK), 0, stream>>>(a1, out + per);
}